// BaseLine_31086973288655
// MI455X (gfx1250) — compile-verified
//
#include <hip/hip_runtime.h>

#define N_NODES   100000
#define N_EDGES   1600000
#define C_IN      64
#define HID       32
#define G_GRAPHS  128
#define NEG_SLOPE 0.01f

typedef __attribute__((ext_vector_type(2))) float v2f;
typedef __attribute__((ext_vector_type(8))) float v8f;

// ---------------- utility ----------------
__global__ void gcn_zero_kernel(float* __restrict__ p, int n) {
  int i = blockIdx.x * blockDim.x + threadIdx.x;
  if (i < n) p[i] = 0.0f;
}

__global__ void gcn_degree_kernel(const int* __restrict__ dst, float* __restrict__ deg) {
  int e = blockIdx.x * blockDim.x + threadIdx.x;
  if (e < N_EDGES) atomicAdd(&deg[dst[e]], 1.0f);
}

__global__ void gcn_counts_kernel(const int* __restrict__ batch, float* __restrict__ cnt) {
  int i = blockIdx.x * blockDim.x + threadIdx.x;
  if (i < N_NODES) atomicAdd(&cnt[batch[i]], 1.0f);
}

__global__ void gcn_dinv_kernel(float* __restrict__ deg) {
  int i = blockIdx.x * blockDim.x + threadIdx.x;
  if (i < N_NODES) deg[i] = rsqrtf(deg[i] + 1.0f);
}

__global__ void gcn_cntinv_kernel(float* __restrict__ cnt) {
  int g = blockIdx.x * blockDim.x + threadIdx.x;
  if (g < G_GRAPHS) cnt[g] = 1.0f / fmaxf(cnt[g], 1.0f);
}

// ---------------- WMMA GEMM: H[n,32] = X[n,K] @ W[K,32] ----------------
// One wave32 computes one 16x16 output tile using V_WMMA_F32_16X16X4_F32.
// ISA 7.12.2 32-bit layouts:
//   A 16x4 : lanes 0-15 -> M=lane, VGPR0=K0, VGPR1=K1 ; lanes 16-31 -> K2,K3
//   B 4x16 : lanes 0-15 -> N=lane, VGPR0=K0, VGPR1=K1 ; lanes 16-31 -> K2,K3
//   C 16x16: VGPR r -> M = r + 8*(lane/16), N = lane%16
template <int K>
__global__ void gcn_gemm_wmma_kernel(const float* __restrict__ X,
                                     const float* __restrict__ W,
                                     float* __restrict__ H, int nRows) {
  const int wave = blockIdx.x * (blockDim.x >> 5) + (threadIdx.x >> 5);
  const int lane = threadIdx.x & 31;
  const int numTiles = (nRows >> 4) * 2;     // (nRows/16) row tiles x 2 col tiles
  if (wave >= numTiles) return;              // wave-uniform: EXEC stays all-ones
  const int rowTile = wave >> 1;
  const int colTile = wave & 1;
  const int half = lane >> 4;                // 0: K pair (0,1); 1: K pair (2,3)
  const int l16  = lane & 15;
  const int aRow = rowTile * 16 + l16;
  const int col  = colTile * 16 + l16;

  v8f c = {};
#pragma unroll
  for (int k0 = 0; k0 < K; k0 += 4) {
    const int ka = k0 + 2 * half;
    v2f a = *(const v2f*)(X + aRow * K + ka);          // 8B-aligned (K%8==0)
    v2f b;
    b.x = W[(ka + 0) * HID + col];
    b.y = W[(ka + 1) * HID + col];
    c = __builtin_amdgcn_wmma_f32_16x16x4_f32(
        /*neg_a=*/false, a, /*neg_b=*/false, b,
        /*c_mod=*/(short)0, c, /*reuse_a=*/false, /*reuse_b=*/false);
  }
#pragma unroll
  for (int r = 0; r < 8; ++r) {
    const int m = r + 8 * half;
    H[(rowTile * 16 + m) * HID + col] = c[r];
  }
}

// ---------------- edge message scatter: agg[dst] += h[src] * dinv[src]*dinv[dst] ----------------
// 8 threads per edge, each moves a float4 (16B gather + 4 f32 atomics into L2-resident agg).
__global__ void gcn_scatter_kernel(const int* __restrict__ src, const int* __restrict__ dst,
                                   const float* __restrict__ dinv,
                                   const float* __restrict__ H, float* __restrict__ agg) {
  int t = blockIdx.x * blockDim.x + threadIdx.x;
  int e = t >> 3;
  if (e >= N_EDGES) return;
  int q = (t & 7) << 2;                       // channel quad start
  int s = src[e], d = dst[e];
  float coef = dinv[s] * dinv[d];
  const float4 hv = *(const float4*)(H + s * HID + q);
  float* ap = agg + d * HID + q;
  atomicAdd(ap + 0, hv.x * coef);
  atomicAdd(ap + 1, hv.y * coef);
  atomicAdd(ap + 2, hv.z * coef);
  atomicAdd(ap + 3, hv.w * coef);
}

// ---------------- combine: self-loop + bias + leaky_relu, fused mean-pool accumulate ----------------
__global__ void gcn_combine_kernel(const float* __restrict__ agg, const float* __restrict__ Hlin,
                                   const float* __restrict__ bias, const float* __restrict__ dinv,
                                   const int* __restrict__ batch, const float* __restrict__ cntinv,
                                   float* __restrict__ Hact, float* __restrict__ feat) {
  int t = blockIdx.x * blockDim.x + threadIdx.x;
  if (t >= N_NODES * HID) return;
  int i = t >> 5, ch = t & 31;
  float di = dinv[i];
  float v = agg[t] + Hlin[t] * di * di + bias[ch];
  v = v > 0.0f ? v : NEG_SLOPE * v;
  Hact[t] = v;
  int g = batch[i];
  atomicAdd(&feat[g * HID + ch], v * cntinv[g]);
}

__global__ void gcn_merge_kernel(const float* __restrict__ feats, float* __restrict__ out) {
  int t = blockIdx.x * blockDim.x + threadIdx.x;
  const int GH = G_GRAPHS * HID;
  if (t < GH) out[t] = (feats[t] + feats[GH + t] + feats[2 * GH + t]) * (1.0f / 3.0f);
}

// ---------------- host side ----------------
static inline int cdiv(int a, int b) { return (a + b - 1) / b; }

extern "C" void kernel_launch(void* const* d_in, const int* in_sizes, int n_in,
                              void* d_out, int out_size, void* d_ws, size_t ws_size,
                              hipStream_t stream) {
  (void)in_sizes; (void)n_in; (void)out_size; (void)ws_size;
  const float* x     = (const float*)d_in[0];
  const float* W0    = (const float*)d_in[1];
  const float* b0    = (const float*)d_in[2];
  const float* W1    = (const float*)d_in[3];
  const float* b1    = (const float*)d_in[4];
  const float* W2    = (const float*)d_in[5];
  const float* b2    = (const float*)d_in[6];
  const int*   src   = (const int*)d_in[7];
  const int*   dst   = (const int*)d_in[8];
  const int*   batch = (const int*)d_in[9];
  float* out = (float*)d_out;

  float* ws    = (float*)d_ws;
  float* deg   = ws;                         // N   (becomes dinv)
  float* cnt   = deg + N_NODES;              // G   (becomes cntinv)
  float* lin   = cnt + G_GRAPHS;             // N*HID  (pre-aggregation h = x@W)
  float* act   = lin + (size_t)N_NODES * HID;// N*HID  (post-activation features)
  float* agg   = act + (size_t)N_NODES * HID;// N*HID  (scatter accumulator)
  float* feats = agg + (size_t)N_NODES * HID;// 3*G*HID (pooled per layer)

  const int T = 256;
  const int nTiles = (N_NODES / 16) * 2;

  // init
  gcn_zero_kernel<<<cdiv(N_NODES, T), T, 0, stream>>>(deg, N_NODES);
  gcn_zero_kernel<<<1, T, 0, stream>>>(cnt, G_GRAPHS);
  gcn_zero_kernel<<<cdiv(3 * G_GRAPHS * HID, T), T, 0, stream>>>(feats, 3 * G_GRAPHS * HID);
  gcn_degree_kernel<<<cdiv(N_EDGES, T), T, 0, stream>>>(dst, deg);
  gcn_counts_kernel<<<cdiv(N_NODES, T), T, 0, stream>>>(batch, cnt);
  gcn_dinv_kernel<<<cdiv(N_NODES, T), T, 0, stream>>>(deg);
  gcn_cntinv_kernel<<<1, T, 0, stream>>>(cnt);

  const float* Ws[3] = {W0, W1, W2};
  const float* bs[3] = {b0, b1, b2};
  const float* hin = x;
  for (int layer = 0; layer < 3; ++layer) {
    if (layer == 0)
      gcn_gemm_wmma_kernel<C_IN><<<cdiv(nTiles, T / 32), T, 0, stream>>>(hin, Ws[0], lin, N_NODES);
    else
      gcn_gemm_wmma_kernel<HID><<<cdiv(nTiles, T / 32), T, 0, stream>>>(hin, Ws[layer], lin, N_NODES);
    gcn_zero_kernel<<<cdiv(N_NODES * HID, T), T, 0, stream>>>(agg, N_NODES * HID);
    gcn_scatter_kernel<<<cdiv(N_EDGES * 8, T), T, 0, stream>>>(src, dst, deg, lin, agg);
    gcn_combine_kernel<<<cdiv(N_NODES * HID, T), T, 0, stream>>>(
        agg, lin, bs[layer], deg, batch, cnt, act, feats + layer * G_GRAPHS * HID);
    hin = act;
  }
  gcn_merge_kernel<<<cdiv(G_GRAPHS * HID, T), T, 0, stream>>>(feats, out);
}